// EBM_64759516889096
// MI455X (gfx1250) — compile-verified
//
#include <hip/hip_runtime.h>
#include <hip/hip_bf16.h>
#include <stdint.h>

// system2_refine: `steps` iterations of softmax-gradient refinement over
// V=50257 vocab. One row (b,t) per workgroup; the 201KB row is staged ONCE
// into the 320KB WGP LDS via CDNA5 async global->LDS DMA (ASYNCcnt path);
// logits + exp values live in registers across all steps. HBM traffic is the
// 824MB read-once/write-once floor (~35us @ 23.3 TB/s).
//
// Mean-centering is deferred to the end: softmax (and thus grad/clip) is
// shift-invariant, and by induction l_s = L_s - mean(L_s) where L is the
// never-centered state, so a single final centering is exactly equivalent.

#define VOCAB 50257
#define BLK   1024
#define NK    ((VOCAB + BLK - 1) / BLK)   // 50 elements per thread
#define ALPHA 0.1f
#define CLAMP 5.0f

// ---- wave32 butterfly reductions -------------------------------------------
__device__ __forceinline__ float wred_sum(float v) {
#pragma unroll
  for (int o = 16; o > 0; o >>= 1) v += __shfl_xor(v, o, 32);
  return v;
}
__device__ __forceinline__ float wred_max(float v) {
#pragma unroll
  for (int o = 16; o > 0; o >>= 1) v = fmaxf(v, __shfl_xor(v, o, 32));
  return v;
}

// ---- block reductions: exactly 32 waves (BLK=1024, wave32) -----------------
__device__ __forceinline__ float bred_max(float v, float* red) {
  const int lane = threadIdx.x & 31, wv = threadIdx.x >> 5;
  v = wred_max(v);
  if (lane == 0) red[wv] = v;
  __syncthreads();
  float r = wred_max(red[lane]);   // every wave reduces all 32 partials
  __syncthreads();                 // red[] reusable afterwards
  return r;
}
__device__ __forceinline__ float bred_sum(float v, float* red) {
  const int lane = threadIdx.x & 31, wv = threadIdx.x >> 5;
  v = wred_sum(v);
  if (lane == 0) red[wv] = v;
  __syncthreads();
  float r = wred_sum(red[lane]);
  __syncthreads();
  return r;
}
__device__ __forceinline__ void bred_sum2(float& a, float& b, float* red) {
  const int lane = threadIdx.x & 31, wv = threadIdx.x >> 5;
  a = wred_sum(a);
  b = wred_sum(b);
  if (lane == 0) { red[wv] = a; red[32 + wv] = b; }
  __syncthreads();
  float ra = wred_sum(red[lane]);
  float rb = wred_sum(red[32 + lane]);
  __syncthreads();
  a = ra; b = rb;
}

// ---- CDNA5 async global -> LDS copy (ASYNCcnt path) ------------------------
// b32 granularity: lane-contiguous dwords coalesce into full cachelines in the
// async engine, and 4B payloads are legal at ANY row base alignment
// (V=50257 is odd, so 3 of 4 rows are only 4B-aligned -> b128 would risk
//  MEMVIOL outside UNALIGNED mode).
__device__ __forceinline__ void async_g2l_b32(uint32_t lds_addr, uint32_t goff,
                                              const void* sbase) {
  asm volatile("global_load_async_to_lds_b32 %0, %1, %2"
               :: "v"(lds_addr), "v"(goff), "s"(sbase) : "memory");
}
__device__ __forceinline__ void wait_async0() {
  asm volatile("s_wait_asynccnt 0" ::: "memory");
}

__global__ void __launch_bounds__(BLK, 1)
ebm_refine_kernel(const float* __restrict__ energies,
                  const int* __restrict__ steps_ptr,
                  float* __restrict__ out, int rows) {
  extern __shared__ __align__(16) float sE[];  // VOCAB floats (201,028 B)
  __shared__ float red[64];                    // cross-wave reduction scratch

  const int t   = threadIdx.x;
  const int row = blockIdx.x;
  const float* __restrict__ gE = energies + (size_t)row * VOCAB;

  // ---- stage one vocab row into LDS with async DMA -------------------------
  {
    const uint32_t ldsBase = (uint32_t)(uintptr_t)(void*)sE;  // LDS byte offset
#pragma unroll 5
    for (int d = t; d < VOCAB; d += BLK) {
      const uint32_t off = (uint32_t)d * 4u;
      async_g2l_b32(ldsBase + off, off, (const void*)gE);
    }
    wait_async0();          // this wave's asyncs done ...
  }
  __syncthreads();          // ... barrier covers all 32 waves

  // ---- logits live in registers: logits0 = -E ------------------------------
  float l[NK];
#pragma unroll
  for (int k = 0; k < NK; ++k) {
    const int idx = t + k * BLK;
    l[k] = (idx < VOCAB) ? -sE[idx] : 0.0f;
  }

  const int   nsteps = steps_ptr[0];
  const float inv_bt = 1.0f / (float)rows;

  for (int s = 0; s < nsteps; ++s) {
    // 1) row max (softmax stability; shift-invariant wrt deferred centering)
    float m = -3.402823466e38f;
#pragma unroll
    for (int k = 0; k < NK; ++k) {
      const int idx = t + k * BLK;
      if (idx < VOCAB) m = fmaxf(m, l[k]);
    }
    m = bred_max(m, red);

    // 2) s0 = sum exp(l-m), s1 = sum exp(l-m)*E   (single fused pass)
    float tk[NK];
    float s0 = 0.0f, s1 = 0.0f;
#pragma unroll
    for (int k = 0; k < NK; ++k) {
      const int idx = t + k * BLK;
      if (idx < VOCAB) {
        const float e = __expf(l[k] - m);     // v_exp_f32
        tk[k] = e;
        s0 += e;
        s1 += e * sE[idx];
      } else {
        tk[k] = 0.0f;
      }
    }
    bred_sum2(s0, s1, red);
    const float is0 = 1.0f / s0;
    const float ee  = s1 * is0;               // E_p[E] for this row

    // 3) l -= alpha * clip(p*(E-ee)*inv_bt)
#pragma unroll
    for (int k = 0; k < NK; ++k) {
      const int idx = t + k * BLK;
      if (idx < VOCAB) {
        const float p = tk[k] * is0;
        float g = p * (sE[idx] - ee) * inv_bt;
        g = fminf(fmaxf(g, -CLAMP), CLAMP);
        l[k] -= ALPHA * g;
      }
    }
  }

  // ---- single deferred mean-centering (exactly equivalent, see header) -----
  float ls = 0.0f;
#pragma unroll
  for (int k = 0; k < NK; ++k) {
    const int idx = t + k * BLK;
    if (idx < VOCAB) ls += l[k];
  }
  ls = bred_sum(ls, red);
  const float mean = ls * (1.0f / (float)VOCAB);

  // ---- write refined logits (coalesced b32 streams) ------------------------
  float* __restrict__ gO = out + (size_t)row * VOCAB;
#pragma unroll
  for (int k = 0; k < NK; ++k) {
    const int idx = t + k * BLK;
    if (idx < VOCAB) gO[idx] = l[k] - mean;
  }
}

extern "C" void kernel_launch(void* const* d_in, const int* in_sizes, int n_in,
                              void* d_out, int out_size, void* d_ws, size_t ws_size,
                              hipStream_t stream) {
  (void)n_in; (void)d_ws; (void)ws_size; (void)out_size;
  const float* energies = (const float*)d_in[0];
  const int*   steps    = (const int*)d_in[1];
  float*       out      = (float*)d_out;

  const int rows = in_sizes[0] / VOCAB;                // B*T = 2048
  const size_t shmem = (size_t)VOCAB * sizeof(float);  // 201,028 B dynamic LDS

  // Opt in to >64KB dynamic LDS (best-effort / no-op on AMD; deterministic).
  (void)hipFuncSetAttribute((const void*)ebm_refine_kernel,
                            hipFuncAttributeMaxDynamicSharedMemorySize,
                            (int)shmem);

  ebm_refine_kernel<<<rows, BLK, shmem, stream>>>(energies, steps, out, rows);
}